// PointChargeField_44839458570525
// MI455X (gfx1250) — compile-verified
//
#include <hip/hip_runtime.h>

// PointChargeField on MI455X (gfx1250, wave32).
// out[b,n] = sum_m q_m * (p_n - c_m) / |p_n - c_m|^3
// Refactored: out_n = p_n * S_n - T_n,  S_n = sum_m w, T_n = sum_m w*c_m,
//             w = q_m * d^-3,  d^2 computed 16x16 at a time with
//             V_WMMA_F32_16X16X4_F32 via d^2 = |p|^2 - 2 p.c + |c|^2.

typedef float v2f __attribute__((ext_vector_type(2)));
typedef float v8f __attribute__((ext_vector_type(8)));

#define B_BATCH 8
#define N_ATOMS 2048
#define M_CHARGES 1024
#define BLOCKS_PER_BATCH 16   // 8 waves/block * 16 atoms/wave = 128 atoms/block
#define THREADS 256

__global__ __launch_bounds__(THREADS)
void PointChargeField_kernel(const float* __restrict__ pos,       // [B,N,3]
                             const float* __restrict__ chg,       // [B,M]
                             const float* __restrict__ cpos,      // [B,M,3]
                             float* __restrict__ out)             // [B,N,3]
{
    // Per charge m: {-2x, -2y, -2z, |c|^2, x, y, z, q}  -> 32 KB (of 320 KB/WGP)
    __shared__ float lds[M_CHARGES * 8];

    const int b       = blockIdx.x / BLOCKS_PER_BATCH;
    const int tileBlk = blockIdx.x % BLOCKS_PER_BATCH;

    const float* cp = cpos + (size_t)b * M_CHARGES * 3;
    const float* qb = chg  + (size_t)b * M_CHARGES;

    // ---- Stage & pre-transform all charges for this batch into LDS ----
    for (int m = threadIdx.x; m < M_CHARGES; m += THREADS) {
        float x = cp[m * 3 + 0];
        float y = cp[m * 3 + 1];
        float z = cp[m * 3 + 2];
        float c2 = x * x + y * y + z * z;
        float* dst = &lds[m * 8];
        dst[0] = -2.0f * x;
        dst[1] = -2.0f * y;
        dst[2] = -2.0f * z;
        dst[3] = c2;
        dst[4] = x;
        dst[5] = y;
        dst[6] = z;
        dst[7] = qb[m];
    }
    __syncthreads();

    const int wave = threadIdx.x >> 5;          // wave32
    const int lane = threadIdx.x & 31;
    const int lo16 = lane & 15;
    const bool hi  = (lane >= 16);

    // Each wave owns 16 consecutive atoms.
    const int n = tileBlk * 128 + wave * 16 + lo16;
    const float* pb = pos + ((size_t)b * N_ATOMS + n) * 3;
    const float px = pb[0], py = pb[1], pz = pb[2];
    const float pn2 = px * px + py * py + pz * pz;

    // B operand (4x16, K x N): col n = (px, py, pz, 1).
    // Layout: VGPR0 = K0 (lanes 0-15) / K2 (lanes 16-31); VGPR1 = K1 / K3.
    v2f Bm;
    Bm.x = hi ? pz : px;
    Bm.y = hi ? 1.0f : py;

    // C init: |p_n|^2 for column n — identical across all 8 C/D VGPRs.
    v8f Cinit;
#pragma unroll
    for (int i = 0; i < 8; ++i) Cinit[i] = pn2;

    float S = 0.0f, Tx = 0.0f, Ty = 0.0f, Tz = 0.0f;

    for (int m0 = 0; m0 < M_CHARGES; m0 += 16) {
        // A operand (16x4, M x K): row m = (-2cx, -2cy, -2cz, |c|^2).
        // Layout: lane m%16; VGPR0 = K0 / K2, VGPR1 = K1 / K3 per lane half.
        const float* la = &lds[(m0 + lo16) * 8 + (hi ? 2 : 0)];
        v2f Am;
        Am.x = la[0];
        Am.y = la[1];

        // d2[r] = |p_n|^2 - 2 p_n.c_m + |c_m|^2, m = m0 + r + 8*hi, n = lane%16
        v8f d2 = __builtin_amdgcn_wmma_f32_16x16x4_f32(
            /*neg_a=*/false, Am, /*neg_b=*/false, Bm,
            /*c_mod=*/(short)0, Cinit, /*reuse_a=*/false, /*reuse_b=*/false);

        const int mbase = m0 + (hi ? 8 : 0);
#pragma unroll
        for (int r = 0; r < 8; ++r) {
            // Uniform address within each lane-half -> LDS broadcast, no conflicts.
            const float4 cq = *(const float4*)&lds[(mbase + r) * 8 + 4]; // x,y,z,q
            float inv = __builtin_amdgcn_rsqf(d2[r]);     // v_rsq_f32
            float w   = inv * inv * inv;                  // d^-3
            float wq  = w * cq.w;                         // q * d^-3
            S  += wq;
            Tx = fmaf(wq, cq.x, Tx);
            Ty = fmaf(wq, cq.y, Ty);
            Tz = fmaf(wq, cq.z, Tz);
        }
    }

    // Lanes 0-15 covered charge rows r=0..7 of each tile, lanes 16-31 rows 8..15.
    S  += __shfl_xor(S,  16, 32);
    Tx += __shfl_xor(Tx, 16, 32);
    Ty += __shfl_xor(Ty, 16, 32);
    Tz += __shfl_xor(Tz, 16, 32);

    if (!hi) {
        float* o = out + ((size_t)b * N_ATOMS + n) * 3;
        o[0] = fmaf(px, S, -Tx);
        o[1] = fmaf(py, S, -Ty);
        o[2] = fmaf(pz, S, -Tz);
    }
}

extern "C" void kernel_launch(void* const* d_in, const int* in_sizes, int n_in,
                              void* d_out, int out_size, void* d_ws, size_t ws_size,
                              hipStream_t stream) {
    const float* positions        = (const float*)d_in[0]; // [8,2048,3]
    const float* charges          = (const float*)d_in[1]; // [8,1024]
    const float* charge_positions = (const float*)d_in[2]; // [8,1024,3]
    float* out = (float*)d_out;                            // [8,2048,3]

    dim3 grid(B_BATCH * BLOCKS_PER_BATCH);
    dim3 block(THREADS);
    PointChargeField_kernel<<<grid, block, 0, stream>>>(
        positions, charges, charge_positions, out);
}